// MatryoshkaHarmonicMixing_53824530154176
// MI455X (gfx1250) — compile-verified
//
#include <hip/hip_runtime.h>
#include <math.h>
#include <stdint.h>

// MatryoshkaHarmonicMixing on MI455X (gfx1250).
//
// out[j] = x[j] + sum_o sigmoid(w[o]) * x[j>>o]  for the (row-local) targets
//   o=1: j in [512,1024), j%2==0
//   o=2: j in [512,2048), j%4==0
//   o=3: j in [512,2048), j%8==0
//
// Strategy: one 2048-float row per block. Stage the row into LDS via CDNA5
// async global->LDS B128 copies (ASYNCcnt / s_wait_asynccnt), then serve the
// identity read AND all three gathers from LDS; single coalesced
// global_store_b128 stream out. HBM traffic = 2 * 134 MB (roofline minimum,
// ~11.5 us at 23.3 TB/s).

#define DMODEL 2048

__global__ __launch_bounds__(256) void mhm_kernel(
    const float* __restrict__ x,
    const float* __restrict__ w,
    float* __restrict__ out)
{
    __shared__ float srow[DMODEL];

    const int row = blockIdx.x;
    const int t   = threadIdx.x;              // 0..255
    const float* __restrict__ xr = x + (size_t)row * DMODEL;   // uniform -> SGPR pair
    float* __restrict__ outr     = out + (size_t)row * DMODEL;

    // ---- async stage: row -> LDS, 2 x b128 per thread (8 KB total) ----
    // LDS byte address = low 32 bits of the flat shared-aperture address.
    uint32_t lds_base = (uint32_t)(uintptr_t)&srow[0];
    uint32_t off0 = (uint32_t)t * 16u;
    uint32_t off1 = off0 + 4096u;

    asm volatile("global_load_async_to_lds_b128 %0, %1, %2"
                 :
                 : "v"(lds_base + off0), "v"(off0), "s"(xr)
                 : "memory");
    asm volatile("global_load_async_to_lds_b128 %0, %1, %2"
                 :
                 : "v"(lds_base + off1), "v"(off1), "s"(xr)
                 : "memory");

    // Per-wave: drain this wave's async copies, then workgroup barrier so
    // every wave sees the full row in LDS.
    asm volatile("s_wait_asynccnt 0" ::: "memory");
    __syncthreads();

    // ---- uniform weights: sigmoid via v_exp + v_rcp (no IEEE div sequence) ----
    const float w1 = __builtin_amdgcn_rcpf(1.0f + __expf(-w[0]));
    const float w2 = __builtin_amdgcn_rcpf(1.0f + __expf(-w[1]));
    const float w3 = __builtin_amdgcn_rcpf(1.0f + __expf(-w[2]));

    // ---- compute two aligned float4 groups per thread ----
#pragma unroll
    for (int half = 0; half < 2; ++half) {
        const int j0 = half * 1024 + t * 4;   // 4-aligned group base

        float4 v = *reinterpret_cast<const float4*>(&srow[j0]);  // ds_load_b128

        if (j0 >= 512) {
            // octave 2 (stride 4): group base always divisible by 4
            v.x += w2 * srow[j0 >> 2];
            // octave 3 (stride 8)
            if ((j0 & 7) == 0)
                v.x += w3 * srow[j0 >> 3];
            // octave 1 (stride 2): targets < 1024 only; lanes j0 and j0+2
            if (j0 < 1024) {
                const int h = j0 >> 1;        // in [256, 512)
                v.x += w1 * srow[h];
                v.z += w1 * srow[h + 1];
            }
        }

        *reinterpret_cast<float4*>(&outr[j0]) = v;  // global_store_b128
    }
}

extern "C" void kernel_launch(void* const* d_in, const int* in_sizes, int n_in,
                              void* d_out, int out_size, void* d_ws, size_t ws_size,
                              hipStream_t stream)
{
    const float* x = (const float*)d_in[0];   // (4, 4096, 2048) fp32
    const float* w = (const float*)d_in[1];   // (3,) fp32
    float* out     = (float*)d_out;

    const int nrows = in_sizes[0] / DMODEL;   // 16384

    mhm_kernel<<<dim3(nrows), dim3(256), 0, stream>>>(x, w, out);
}